// ScatterND_22316650070071
// MI455X (gfx1250) — compile-verified
//
#include <hip/hip_runtime.h>
#include <stdint.h>

// ============================================================================
// ScatterND row-set: out = data; out[idx[i]] = updates[i]  (last i wins)
// data: (R=100000, C=512) f32, indices: (U=50000) i32, updates: (U, C) f32
//
// Memory-bound: ~410 MB min traffic -> ~17.6 us at 23.3 TB/s.
// Strategy: resolve winning source per row (owner array), then write every
// output row exactly once. Bulk 2KB-row copy uses the CDNA5 async
// global<->LDS DMA path (ASYNCcnt) so the streaming data never touches VGPRs.
// ============================================================================

#define COLS 512          // floats per row (2048 bytes)
#define WPB  8            // waves per block (256 threads, wave32)
#define F4_PER_ROW (COLS / 4)   // 128 float4 per row
#define CHUNKS (F4_PER_ROW / 32) // 4 async b128 per lane per row

__global__ __launch_bounds__(256) void init_owner_kernel(int* __restrict__ owner,
                                                         int nrows) {
    int i = blockIdx.x * blockDim.x + threadIdx.x;
    if (i < nrows) owner[i] = -1;
}

__global__ __launch_bounds__(256) void build_owner_kernel(const int* __restrict__ idx,
                                                          int* __restrict__ owner,
                                                          int nupd, int nrows) {
    int i = blockIdx.x * blockDim.x + threadIdx.x;
    if (i < nupd) {
        int r = idx[i];
        if (r >= 0 && r < nrows) {
            atomicMax(&owner[r], i);   // last update index wins (deterministic)
        }
    }
}

// One wave32 per output row. Row copy staged through a private 2KB LDS slice
// using gfx1250 async LDS DMA: 4x global_load_async_to_lds_b128 (512B per
// instruction across 32 lanes), s_wait_asynccnt 0, then 4x
// global_store_async_from_lds_b128. Pending async stores are drained by the
// implicit wait-idle at s_endpgm.
__global__ __launch_bounds__(256) void gather_rows_kernel(
    const float* __restrict__ data,
    const float* __restrict__ updates,
    const int* __restrict__ owner,
    float* __restrict__ out,
    int nrows) {

    __shared__ float4 smem[WPB * F4_PER_ROW];   // 16 KB: 2 KB per wave

    const int lane = threadIdx.x & 31;
    const int wave = threadIdx.x >> 5;
    const int row  = blockIdx.x * WPB + wave;
    if (row >= nrows) return;

    const int w = owner[row];   // same addr across wave -> single broadcast load
    const float* __restrict__ src =
        (w >= 0) ? (updates + (size_t)w * COLS) : (data + (size_t)row * COLS);
    float* __restrict__ dst = out + (size_t)row * COLS;

    // LDS byte offset of this wave's slice. Flat LDS aperture keeps the LDS
    // byte address in addr[31:0], so truncation yields the DS offset.
    const uint32_t lds_base = (uint32_t)(uintptr_t)(&smem[wave * F4_PER_ROW]);

    // ---- global -> LDS (async, ASYNCcnt) ----
#pragma unroll
    for (int c = 0; c < CHUNKS; ++c) {
        const int e = c * 32 + lane;                      // float4 index in row
        uint32_t laddr = lds_base + (uint32_t)e * 16u;
        uint64_t gaddr = (uint64_t)(uintptr_t)(src + (size_t)e * 4);
        asm volatile("global_load_async_to_lds_b128 %0, %1, off"
                     :
                     : "v"(laddr), "v"(gaddr)
                     : "memory");
    }

    // RAW on the LDS slice: wait for the async loads before async stores read it.
    asm volatile("s_wait_asynccnt 0" ::: "memory");

    // ---- LDS -> global (async, ASYNCcnt) ----
#pragma unroll
    for (int c = 0; c < CHUNKS; ++c) {
        const int e = c * 32 + lane;
        uint32_t laddr = lds_base + (uint32_t)e * 16u;
        uint64_t gaddr = (uint64_t)(uintptr_t)(dst + (size_t)e * 4);
        asm volatile("global_store_async_from_lds_b128 %0, %1, off"
                     :
                     : "v"(gaddr), "v"(laddr)
                     : "memory");
    }
    // s_endpgm performs an implicit wait-idle, draining outstanding async stores.
}

extern "C" void kernel_launch(void* const* d_in, const int* in_sizes, int n_in,
                              void* d_out, int out_size, void* d_ws, size_t ws_size,
                              hipStream_t stream) {
    const float* data    = (const float*)d_in[0];
    const int*   indices = (const int*)d_in[1];
    const float* updates = (const float*)d_in[2];
    float* out = (float*)d_out;

    const int nupd  = in_sizes[1];           // 50000
    const int nrows = in_sizes[0] / COLS;    // 100000

    int* owner = (int*)d_ws;                 // nrows * 4 bytes of scratch

    // 1) owner[r] = -1
    {
        int threads = 256;
        int blocks  = (nrows + threads - 1) / threads;
        init_owner_kernel<<<blocks, threads, 0, stream>>>(owner, nrows);
    }
    // 2) owner[idx[i]] = max(i)  -> last update wins
    {
        int threads = 256;
        int blocks  = (nupd + threads - 1) / threads;
        build_owner_kernel<<<blocks, threads, 0, stream>>>(indices, owner, nupd, nrows);
    }
    // 3) write every output row once, from updates[owner] or data
    {
        int threads = 256;                   // 8 waves -> 8 rows per block
        int blocks  = (nrows + WPB - 1) / WPB;
        gather_rows_kernel<<<blocks, threads, 0, stream>>>(data, updates, owner, out, nrows);
    }
}